// masked_NLLLoss_37718402793473
// MI455X (gfx1250) — compile-verified
//
#include <hip/hip_runtime.h>

// masked NLL gather:  out[i] = -cost[i * V + inputs[i]] * mask[i]
// N = 16384, V = 32000 (read from in_sizes at launch).
//
// Latency-bound scattered gather; no matrix contraction -> WMMA inapplicable.
// Strategy:
//  - 4 rows per thread: one b128 index load, one b128 mask load, four
//    independent b32 gathers (128 outstanding gathers per wave32 behind a
//    single wait), one b128 store.
//  - All traffic is single-use -> non-temporal hints (th:NT) so the gathered
//    cache lines (128B fetched, 4B used) don't occupy L2.
//  - 32-bit gather offsets: max element index (n-1)*v + (v-1) < 2^29, byte
//    offset < 2^31, so unsigned 32-bit math is exact and the compiler can use
//    SGPR-base + 32-bit VGPR-offset global addressing.
//  - Vec4 kernel assumes n % 4 == 0 (true for N=16384); scalar fallback kernel
//    selected at launch otherwise, so the hot kernel has no tail control flow.

// Clang ext_vector types: required by __builtin_nontemporal_load/store
// (HIP's int4/float4 are structs and are rejected).
typedef int   v4i __attribute__((ext_vector_type(4)));
typedef float v4f __attribute__((ext_vector_type(4)));

__global__ __launch_bounds__(256) void masked_nll_vec4_kernel(
    const float* __restrict__ cost,
    const int*   __restrict__ inputs,
    const float* __restrict__ mask,
    float*       __restrict__ out,
    int nvec,            // n / 4
    unsigned v)
{
    const int t = blockIdx.x * blockDim.x + threadIdx.x;
    if (t >= nvec) return;
    const int base = t * 4;

    const v4i idx4 = __builtin_nontemporal_load(
                         reinterpret_cast<const v4i*>(inputs + base));
    const v4f m4   = __builtin_nontemporal_load(
                         reinterpret_cast<const v4f*>(mask + base));

    const unsigned o0 = (unsigned)(base + 0) * v + (unsigned)idx4.x;
    const unsigned o1 = (unsigned)(base + 1) * v + (unsigned)idx4.y;
    const unsigned o2 = (unsigned)(base + 2) * v + (unsigned)idx4.z;
    const unsigned o3 = (unsigned)(base + 3) * v + (unsigned)idx4.w;

    // Four independent scattered gathers, non-temporal (lines are dead after).
    const float c0 = __builtin_nontemporal_load(cost + o0);
    const float c1 = __builtin_nontemporal_load(cost + o1);
    const float c2 = __builtin_nontemporal_load(cost + o2);
    const float c3 = __builtin_nontemporal_load(cost + o3);

    v4f r;
    r.x = -c0 * m4.x;
    r.y = -c1 * m4.y;
    r.z = -c2 * m4.z;
    r.w = -c3 * m4.w;
    __builtin_nontemporal_store(r, reinterpret_cast<v4f*>(out + base));
}

// General fallback (1 row / thread), only launched when n % 4 != 0.
__global__ __launch_bounds__(256) void masked_nll_scalar_kernel(
    const float* __restrict__ cost,
    const int*   __restrict__ inputs,
    const float* __restrict__ mask,
    float*       __restrict__ out,
    int n, unsigned v)
{
    const int i = blockIdx.x * blockDim.x + threadIdx.x;
    if (i >= n) return;
    const unsigned o = (unsigned)i * v + (unsigned)inputs[i];
    const float c = __builtin_nontemporal_load(cost + o);
    out[i] = -c * mask[i];
}

extern "C" void kernel_launch(void* const* d_in, const int* in_sizes, int n_in,
                              void* d_out, int out_size, void* d_ws, size_t ws_size,
                              hipStream_t stream)
{
    (void)n_in; (void)out_size; (void)d_ws; (void)ws_size;

    const float* cost   = (const float*)d_in[0];
    const int*   inputs = (const int*)  d_in[1];   // integer input -> const int*
    const float* mask   = (const float*)d_in[2];
    float*       out    = (float*)d_out;

    const int n = in_sizes[1];                 // 16384 rows (len of inputs/mask)
    const unsigned v = (unsigned)(in_sizes[0] / n);  // 32000 vocab

    const int threads = 256;                   // 8 wave32s per block
    if ((n & 3) == 0) {
        const int nvec   = n / 4;
        const int blocks = (nvec + threads - 1) / threads;
        masked_nll_vec4_kernel<<<blocks, threads, 0, stream>>>(
            cost, inputs, mask, out, nvec, v);
    } else {
        const int blocks = (n + threads - 1) / threads;
        masked_nll_scalar_kernel<<<blocks, threads, 0, stream>>>(
            cost, inputs, mask, out, n, v);
    }
}